// Attention_60017872994742
// MI455X (gfx1250) — compile-verified
//
#include <hip/hip_runtime.h>
#include <hip/hip_bf16.h>

// Additive (Bahdanau) attention for MI455X / gfx1250.
// B=4, QL=256, KL=1024, QD=256, KD=512, VD=512.
//
// Roofline: all intermediates (~10 MB) live in the 192 MB L2; at 23.3 TB/s
// HBM is irrelevant. The dominant cost is 268M tanh evaluations (TRANS-bound),
// so the score stage uses v_tanh_f32 and single-b128-per-row-per-thread loads,
// staged through LDS with GLOBAL_LOAD_ASYNC_TO_LDS_B128 double buffering
// (ASYNCcnt). The three true GEMMs use V_WMMA_F32_16X16X4_F32.

typedef __attribute__((ext_vector_type(2))) float v2f;
typedef __attribute__((ext_vector_type(4))) float v4f;
typedef __attribute__((ext_vector_type(8))) float v8f;
typedef __attribute__((ext_vector_type(4))) int   v4i;
typedef __attribute__((ext_vector_type(4))) unsigned char v4u8;

#define B_   4
#define QL_  256
#define KL_  1024
#define QD_  256
#define KD_  512
#define VD_  512
#define NEG_INF_ (-1.0e12f)

__device__ __forceinline__ float fast_tanh(float x) {
#if defined(__gfx1250__) && __has_builtin(__builtin_amdgcn_tanhf)
    return __builtin_amdgcn_tanhf(x);      // v_tanh_f32 (gfx1250 TRANS op)
#else
    float ex = __expf(2.0f * x);           // v_exp_f32
    return 1.0f - 2.0f * __builtin_amdgcn_rcpf(ex + 1.0f);
#endif
}

// ---- gfx1250 async global->LDS path (guarded; fallback = direct b128) ----
// Builtin prototype (from compiler diagnostic): (v4i as1*, v4i as3*, Ii, Ii)
#if defined(__gfx1250__) && __has_builtin(__builtin_amdgcn_global_load_async_to_lds_b128)
#define HAVE_ASYNC_LDS 1
typedef __attribute__((address_space(1))) v4i as1_v4i;
typedef __attribute__((address_space(3))) v4i as3_v4i;
__device__ __forceinline__ void async_copy16(const void* gsrc, void* ldst) {
    // generic->AS1 is numerically identity; generic LDS low 32 bits = LDS offset
    as1_v4i* g = (as1_v4i*)(unsigned long long)(__SIZE_TYPE__)gsrc;
    as3_v4i* l = (as3_v4i*)(unsigned int)(__SIZE_TYPE__)ldst;
    __builtin_amdgcn_global_load_async_to_lds_b128(g, l, 0, 0);
}
__device__ __forceinline__ void wait_async_le1() {
#if __has_builtin(__builtin_amdgcn_s_wait_asynccnt)
    __builtin_amdgcn_s_wait_asynccnt(1);
#else
    asm volatile("s_wait_asynccnt 0x1" ::: "memory");
#endif
}
__device__ __forceinline__ void wait_async_le0() {
#if __has_builtin(__builtin_amdgcn_s_wait_asynccnt)
    __builtin_amdgcn_s_wait_asynccnt(0);
#else
    asm volatile("s_wait_asynccnt 0x0" ::: "memory");
#endif
}
#else
#define HAVE_ASYNC_LDS 0
#endif

// ---------------------------------------------------------------------------
// GEMM 1: mQ[m][n] = sum_d Q[m][d] * Wq[n][d] + bq[n]
//         M = B*QL = 1024, N = QD = 256, K = QD = 256.
// One wave computes one 16x16 D tile; K stepped by 4 per WMMA.
// ---------------------------------------------------------------------------
__global__ __launch_bounds__(256) void k_gemm_mq(
    const float* __restrict__ Q, const float* __restrict__ Wq,
    const float* __restrict__ bq, float* __restrict__ mQ) {
    const int Kdim = QD_;                       // 256
    int gw   = (blockIdx.x * blockDim.x + threadIdx.x) >> 5;
    int lane = threadIdx.x & 31;
    int tn   = gw & 15;                         // 16 N tiles
    int tm   = gw >> 4;                         // 64 M tiles
    int half = lane >> 4, l15 = lane & 15;
    const float* arow = Q  + (size_t)(tm * 16 + l15) * Kdim;
    const float* brow = Wq + (size_t)(tn * 16 + l15) * Kdim;
    v8f acc = {};
    #pragma unroll 4
    for (int k0 = 0; k0 < Kdim; k0 += 4) {
        int kk = k0 + half * 2;
        v2f a = *(const v2f*)(arow + kk);
        v2f b = *(const v2f*)(brow + kk);
        acc = __builtin_amdgcn_wmma_f32_16x16x4_f32(
            false, a, false, b, (short)0, acc, false, false);
    }
    int n = tn * 16 + l15;
    float bias = bq[n];
    int mbase = tm * 16 + half * 8;
    #pragma unroll
    for (int r = 0; r < 8; ++r)
        mQ[(size_t)(mbase + r) * QD_ + n] = acc[r] + bias;
}

// ---------------------------------------------------------------------------
// GEMM 2 (transposed output): mKt[b][e][k] = sum_d Wk[e][d]*K[b][k][d] + bk[e]
//   A = Wk (M = e = 256), B = K rows (N = k = 1024 per batch), Kdim = 512.
// Choosing A=Wk makes D's N axis the k index -> coalesced store of the
// e-major layout the tanh stage wants.
// ---------------------------------------------------------------------------
__global__ __launch_bounds__(256) void k_gemm_mkt(
    const float* __restrict__ Kin, const float* __restrict__ Wk,
    const float* __restrict__ bk, float* __restrict__ mKt) {
    const int Kdim = KD_;                       // 512
    int gw   = (blockIdx.x * blockDim.x + threadIdx.x) >> 5;
    int lane = threadIdx.x & 31;
    int b    = gw >> 10;                        // 1024 tiles per batch
    int rem  = gw & 1023;
    int tm   = rem >> 6;                        // 16 e-tiles
    int tn   = rem & 63;                        // 64 k-tiles
    int half = lane >> 4, l15 = lane & 15;
    const float* arow = Wk  + (size_t)(tm * 16 + l15) * Kdim;
    const float* brow = Kin + ((size_t)b * KL_ + tn * 16 + l15) * Kdim;
    v8f acc = {};
    #pragma unroll 4
    for (int k0 = 0; k0 < Kdim; k0 += 4) {
        int kk = k0 + half * 2;
        v2f a  = *(const v2f*)(arow + kk);
        v2f bb = *(const v2f*)(brow + kk);
        acc = __builtin_amdgcn_wmma_f32_16x16x4_f32(
            false, a, false, bb, (short)0, acc, false, false);
    }
    int n = tn * 16 + l15;
    int mbase = tm * 16 + half * 8;
    float* out = mKt + (size_t)b * QD_ * KL_;
    #pragma unroll
    for (int r = 0; r < 8; ++r) {
        int m = mbase + r;
        out[(size_t)m * KL_ + n] = acc[r] + bk[m];
    }
}

// ---------------------------------------------------------------------------
// Stage 3: scores + mask + softmax -> P (attn_weights, second output slot).
// One 256-thread block per (b,q); thread t owns k = 4t..4t+3 so each mKt row
// (4 KB) is fetched as one coalesced b128 per thread. mq/wV broadcast from
// LDS. Rows are double-buffered via GLOBAL_LOAD_ASYNC_TO_LDS_B128 +
// s_wait_asynccnt (no barrier needed: each wave reads back exactly the bytes
// its own lanes transferred; async loads complete in order).
// Hot loop: 4x v_tanh_f32 per e per thread (the whole problem's bottleneck).
// ---------------------------------------------------------------------------
__global__ __launch_bounds__(256) void k_scores_softmax(
    const float* __restrict__ mQ, const float* __restrict__ mKt,
    const float* __restrict__ wV, const float* __restrict__ bV,
    const unsigned char* __restrict__ mask, float* __restrict__ P) {
    __shared__ float mq_s[QD_];
    __shared__ float wv_s[QD_];
    __shared__ float red[256];
#if HAVE_ASYNC_LDS
    __shared__ float kbuf[2][KL_];              // 8 KB double buffer
#endif
    int bq = blockIdx.x;                        // 0..B*QL-1
    int b  = bq >> 8;
    int t  = threadIdx.x;                       // 0..255
    mq_s[t] = mQ[(size_t)bq * QD_ + t];
    wv_s[t] = wV[t];
    const float* mk = mKt + (size_t)b * QD_ * KL_;
#if HAVE_ASYNC_LDS
    async_copy16(mk + 4 * t, &kbuf[0][4 * t]);  // prime row e=0
#endif
    __syncthreads();                            // mq_s/wv_s visible to all
    float s0 = 0.f, s1 = 0.f, s2 = 0.f, s3 = 0.f;
    #pragma unroll 2
    for (int e = 0; e < QD_; ++e) {
#if HAVE_ASYNC_LDS
        if (e + 1 < QD_) {
            async_copy16(mk + (size_t)(e + 1) * KL_ + 4 * t,
                         &kbuf[(e + 1) & 1][4 * t]);
            wait_async_le1();                   // row e complete (in-order)
        } else {
            wait_async_le0();
        }
        v4f x = *(const v4f*)&kbuf[e & 1][4 * t];
#else
        int ep = (e + 4 < QD_) ? e + 4 : QD_ - 1;
        __builtin_prefetch(mk + (size_t)ep * KL_ + 4 * t, 0, 1);
        v4f x = *(const v4f*)(mk + (size_t)e * KL_ + 4 * t);
#endif
        float w = wv_s[e], a = mq_s[e];
        s0 += w * fast_tanh(a + x[0]);
        s1 += w * fast_tanh(a + x[1]);
        s2 += w * fast_tanh(a + x[2]);
        s3 += w * fast_tanh(a + x[3]);
    }
    float bv = bV[0];
    v4u8 m4 = *(const v4u8*)(mask + (size_t)b * KL_ + 4 * t);
    float v0 = m4[0] ? NEG_INF_ : s0 + bv;
    float v1 = m4[1] ? NEG_INF_ : s1 + bv;
    float v2 = m4[2] ? NEG_INF_ : s2 + bv;
    float v3 = m4[3] ? NEG_INF_ : s3 + bv;
    // block max over 1024 scores
    red[t] = fmaxf(fmaxf(v0, v1), fmaxf(v2, v3));
    __syncthreads();
    for (int off = 128; off > 0; off >>= 1) {
        if (t < off) red[t] = fmaxf(red[t], red[t + off]);
        __syncthreads();
    }
    float mx = red[0];
    __syncthreads();
    float e0 = __expf(v0 - mx), e1 = __expf(v1 - mx);
    float e2 = __expf(v2 - mx), e3 = __expf(v3 - mx);
    red[t] = e0 + e1 + e2 + e3;
    __syncthreads();
    for (int off = 128; off > 0; off >>= 1) {
        if (t < off) red[t] += red[t + off];
        __syncthreads();
    }
    float inv = 1.0f / red[0];
    v4f r = { e0 * inv, e1 * inv, e2 * inv, e3 * inv };
    *(v4f*)(P + (size_t)bq * KL_ + 4 * t) = r;
}

// ---------------------------------------------------------------------------
// GEMM 4: ctx[b][q][v] = sum_k P[b][q][k] * V[b][k][v]
//   M = QL = 256, N = VD = 512, Kdim = KL = 1024, per batch.
// ---------------------------------------------------------------------------
__global__ __launch_bounds__(256) void k_gemm_ctx(
    const float* __restrict__ P, const float* __restrict__ Vin,
    float* __restrict__ ctx) {
    int gw   = (blockIdx.x * blockDim.x + threadIdx.x) >> 5;
    int lane = threadIdx.x & 31;
    int b    = gw >> 9;                         // 512 tiles per batch
    int rem  = gw & 511;
    int tm   = rem >> 5;                        // 16 q-tiles
    int tn   = rem & 31;                        // 32 v-tiles
    int half = lane >> 4, l15 = lane & 15;
    const float* arow = P   + ((size_t)b * QL_ + tm * 16 + l15) * KL_;
    const float* bcol = Vin + (size_t)b * KL_ * VD_ + (tn * 16 + l15);
    v8f acc = {};
    #pragma unroll 4
    for (int k0 = 0; k0 < KL_; k0 += 4) {
        int kk = k0 + half * 2;
        v2f a  = *(const v2f*)(arow + kk);
        v2f bb = { bcol[(size_t)kk * VD_], bcol[(size_t)(kk + 1) * VD_] };
        acc = __builtin_amdgcn_wmma_f32_16x16x4_f32(
            false, a, false, bb, (short)0, acc, false, false);
    }
    int n = tn * 16 + l15;
    int mbase = tm * 16 + half * 8;
    float* out = ctx + (size_t)b * QL_ * VD_;
    #pragma unroll
    for (int r = 0; r < 8; ++r)
        out[(size_t)(mbase + r) * VD_ + n] = acc[r];
}

// ---------------------------------------------------------------------------
extern "C" void kernel_launch(void* const* d_in, const int* in_sizes, int n_in,
                              void* d_out, int out_size, void* d_ws, size_t ws_size,
                              hipStream_t stream) {
    const float*         Q    = (const float*)d_in[0];   // [4,256,256]
    const float*         K    = (const float*)d_in[1];   // [4,1024,512]
    const float*         V    = (const float*)d_in[2];   // [4,1024,512]
    const unsigned char* mask = (const unsigned char*)d_in[3]; // [4,1024] bool
    const float*         Wq   = (const float*)d_in[4];   // [256,256]
    const float*         bq   = (const float*)d_in[5];   // [256]
    const float*         Wk   = (const float*)d_in[6];   // [256,512]
    const float*         bk   = (const float*)d_in[7];   // [256]
    const float*         wV   = (const float*)d_in[8];   // [1,256]
    const float*         bV   = (const float*)d_in[9];   // [1]

    float* ctx = (float*)d_out;                              // [4,256,512]
    float* P   = (float*)d_out + (size_t)B_ * QL_ * VD_;     // [4,256,1024]
    float* mQ  = (float*)d_ws;                               // 1 MB
    float* mKt = (float*)d_ws + (size_t)B_ * QL_ * QD_;      // 4 MB

    // 1024 wave-tiles (64x16), 8 waves/block
    k_gemm_mq <<<dim3(128), dim3(256), 0, stream>>>(Q, Wq, bq, mQ);
    // 4096 wave-tiles (4 x 16 x 64)
    k_gemm_mkt<<<dim3(512), dim3(256), 0, stream>>>(K, Wk, bk, mKt);
    // one block per (b,q)
    k_scores_softmax<<<dim3(B_ * QL_), dim3(256), 0, stream>>>(mQ, mKt, wV, bV, mask, P);
    // 2048 wave-tiles (4 x 16 x 32)
    k_gemm_ctx<<<dim3(256), dim3(256), 0, stream>>>(P, V, ctx);
}